// PopGroup_15444702396967
// MI455X (gfx1250) — compile-verified
//
#include <hip/hip_runtime.h>

// Fused TGN memory update for MI455X (gfx1250, wave32).
//   msgs = X @ W_reduce^T ; gi = msgs @ w_ih^T + b_ih ; gh = h @ w_hh^T + b_hh
//   GRU gates -> h_new ; out = copy(node_memories) with rows[node_ids] = h_new
//
// All GEMMs use V_WMMA_F32_16X16X4_F32 (exact fp32). Per block (8 waves /
// 128 batch rows) weights are staged into LDS once and shared (8x less L2
// traffic than per-wave reads); staging uses GLOBAL_LOAD_ASYNC_TO_LDS_B128
// with double buffering so the L2 pull for stage i+1 hides behind the WMMA
// compute of stage i (one s_wait_asynccnt + one barrier per stage).
// GEMM1 runs k-outer with 11 live accumulators so X is read from HBM once.
// All LDS rows use stride s with (s/4 mod 16) odd -> the 32-lane WMMA
// fragment read covers all 64 banks exactly once (conflict-free).

typedef float v2f __attribute__((ext_vector_type(2)));
typedef float v8f __attribute__((ext_vector_type(8)));

#define MEM_DIM 172
#define RAW_DIM 512
#define GATE3   516              // 3*MEM_DIM
#define PADT    180              // msgs/h tile row stride (52 mod 64, 16B aligned)
#define PADW2   180              // staged gate-weight row stride
#define PADK    68               // staged W_reduce k-chunk row stride (4 mod 64)
#define KC      64               // GEMM1 k-chunk
#define NT1     11               // msgs col tiles (176/16)
#define WAVES_PER_BLK 8
#define NTHREADS (WAVES_PER_BLK * 32)
#define TILE_FLOATS (2 * 16 * PADT)          // msgs + h per wave (5760)
#define WBUF_FLOATS (6 * 16 * PADW2)         // 17280 >= 176*PADK = 11968
#define SMEM_FLOATS (WAVES_PER_BLK * TILE_FLOATS + 2 * WBUF_FLOATS) // 80640 fl

__device__ __forceinline__ float fast_sigmoid(float x) {
    return 1.0f / (1.0f + __expf(-x));
}
__device__ __forceinline__ float fast_tanh(float x) {
    float e = __expf(-2.0f * x);
    return (1.0f - e) / (1.0f + e);
}
__device__ __forceinline__ v8f wmma_f32(v2f a, v2f b, v8f c) {
    // 8 args: (neg_a, A, neg_b, B, c_mod, C, reuse_a, reuse_b)
    return __builtin_amdgcn_wmma_f32_16x16x4_f32(false, a, false, b,
                                                 (short)0, c, false, false);
}

// Async global->LDS copy, 16B per lane; tracked by ASYNCcnt.
// LDS byte offset = low 32 bits of the flat pointer.
__device__ __forceinline__ void async_copy_b128(float* lds_dst, const float* gsrc) {
    asm volatile("global_load_async_to_lds_b128 %0, %1, off"
                 :: "v"((unsigned)(unsigned long long)lds_dst), "v"(gsrc)
                 : "memory");
}
__device__ __forceinline__ void wait_async_all() {
    asm volatile("s_wait_asynccnt 0x0" ::: "memory");
}

// Stage Wr[0:176][kc:kc+KC] -> buf[176][PADK] (rows >= 172 clamped).
__device__ __forceinline__ void stage_wr(float* buf, const float* __restrict__ Wr,
                                         int kc, int tid) {
    for (int t = tid; t < 176 * (KC / 4); t += NTHREADS) {
        const int r = t >> 4;                    // 16 float4 per row
        const int q = (t & 15) << 2;
        const int wrow = (r < MEM_DIM) ? r : (MEM_DIM - 1);
        async_copy_b128(buf + r * PADK + q,
                        Wr + (long long)wrow * RAW_DIM + kc + q);
    }
}

// Stage 6 gate-weight groups [16][172] -> buf[(g*16+r)][PADW2].
__device__ __forceinline__ void stage_gates(float* buf,
                                            const float* __restrict__ w_ih,
                                            const float* __restrict__ w_hh,
                                            int j0, int tid) {
    for (int t = tid; t < 6 * 16 * 43; t += NTHREADS) {
        const int g   = t / (16 * 43);
        const int rem = t - g * (16 * 43);
        const int r   = rem / 43;
        const int q   = (rem - r * 43) << 2;
        int wrow = (g % 3) * MEM_DIM + j0 + r;
        if (wrow > GATE3 - 1) wrow = GATE3 - 1;
        const float* W = (g < 3) ? w_ih : w_hh;
        async_copy_b128(buf + (g * 16 + r) * PADW2 + q,
                        W + (long long)wrow * MEM_DIM + q);
    }
}

__global__ __launch_bounds__(NTHREADS)
void tgn_gru_fused(const float* __restrict__ node_memories,
                   const float* __restrict__ X,        // [B, 512]
                   const float* __restrict__ Wr,       // [172, 512]
                   const float* __restrict__ w_ih,     // [516, 172]
                   const float* __restrict__ w_hh,     // [516, 172]
                   const float* __restrict__ b_ih,     // [516]
                   const float* __restrict__ b_hh,     // [516]
                   const int*   __restrict__ node_ids, // [B] int32
                   float* __restrict__ out,            // [N, 172]
                   int batch)
{
    extern __shared__ float smem[];
    const int tid       = threadIdx.x;
    const int lane      = tid & 31;
    const int waveInBlk = tid >> 5;
    const int ntiles    = (batch + 15) >> 4;

    int tile = blockIdx.x * WAVES_PER_BLK + waveInBlk;
    const bool wactive = (tile < ntiles);
    if (!wactive) tile = ntiles - 1;        // keep wave alive for barriers
    const int rb0 = tile * 16;

    float* msgsT = smem + waveInBlk * TILE_FLOATS;     // [16][PADT]
    float* hT    = msgsT + 16 * PADT;                  // [16][PADT]
    float* wb0   = smem + WAVES_PER_BLK * TILE_FLOATS; // weight buffer 0
    float* wb1   = wb0 + WBUF_FLOATS;                  // weight buffer 1

    const int ln = lane & 15;       // row/col-within-16 for this lane
    const int hl = lane >> 4;       // half-wave select
    const int c  = hl * 2;          // K sub-offset of A/B fragments

    // ---- async-gather per-node memories h into LDS (completes at 1st wait)
    for (int t = lane; t < 16 * 43; t += 32) {
        const int r = t / 43;
        const int q = (t - r * 43) << 2;
        int rr = rb0 + r; if (rr >= batch) rr = batch - 1;
        const long long nid = (long long)node_ids[rr];
        async_copy_b128(hT + r * PADT + q,
                        node_memories + nid * (long long)MEM_DIM + q);
    }

    // ---- GEMM1: msgs[16,176] = X_tile @ Wr^T, k-outer, 11 live accs ----
    v8f acc[NT1];
    #pragma unroll
    for (int t = 0; t < NT1; ++t) acc[t] = (v8f){0.f,0.f,0.f,0.f,0.f,0.f,0.f,0.f};

    int arow = rb0 + ln; if (arow >= batch) arow = batch - 1;
    const float* aBase = X + (long long)arow * RAW_DIM + c;

    stage_wr(wb0, Wr, 0, tid);      // prologue stage (async)
    wait_async_all();
    __syncthreads();

    for (int i = 0; i < RAW_DIM / KC; ++i) {
        const int kc = i * KC;
        if (kc + KC < RAW_DIM)      // overlap: stage next chunk while computing
            stage_wr((i & 1) ? wb0 : wb1, Wr, kc + KC, tid);
        const float* cw = (i & 1) ? wb1 : wb0;
        #pragma unroll 2
        for (int k0 = 0; k0 < KC; k0 += 4) {
            const v2f a = *(const v2f*)(aBase + kc + k0);   // X read once
            #pragma unroll
            for (int t = 0; t < NT1; ++t) {
                const v2f b = *(const v2f*)(cw + (t * 16 + ln) * PADK + k0 + c);
                acc[t] = wmma_f32(a, b, acc[t]);
            }
        }
        wait_async_all();           // own async deposits done
        __syncthreads();            // everyone's deposits visible / reads done
    }
    // D tiles (row = r + 8*hl, col = 16t+ln) -> LDS (transpose for GEMM2 A)
    #pragma unroll
    for (int t = 0; t < NT1; ++t) {
        #pragma unroll
        for (int r = 0; r < 8; ++r)
            msgsT[(r + hl * 8) * PADT + t * 16 + ln] = acc[t][r];
    }

    // ---- GEMM2/3 + GRU + scatter, per 16-wide output col tile ----
    const float* amBase = msgsT + ln * PADT + c;   // A (msgs) from LDS
    const float* ahBase = hT    + ln * PADT + c;   // A (h)    from LDS

    stage_gates(wb0, w_ih, w_hh, 0, tid);          // prologue stage (async)
    wait_async_all();
    __syncthreads();

    for (int j = 0; j < NT1; ++j) {
        const int j0 = j * 16;
        if (j0 + 16 < MEM_DIM)      // overlap: stage next col tile's weights
            stage_gates((j & 1) ? wb0 : wb1, w_ih, w_hh, j0 + 16, tid);
        const float* cw = (j & 1) ? wb1 : wb0;

        const float* pir = cw + (0 * 16 + ln) * PADW2 + c;
        const float* piz = cw + (1 * 16 + ln) * PADW2 + c;
        const float* pin = cw + (2 * 16 + ln) * PADW2 + c;
        const float* phr = cw + (3 * 16 + ln) * PADW2 + c;
        const float* phz = cw + (4 * 16 + ln) * PADW2 + c;
        const float* phn = cw + (5 * 16 + ln) * PADW2 + c;

        const int jr = j0 + ln;
        const int jz = MEM_DIM + j0 + ln;
        int jn = 2 * MEM_DIM + j0 + ln; if (jn > GATE3 - 1) jn = GATE3 - 1;

        const float bir = b_ih[jr], biz = b_ih[jz], bin_ = b_ih[jn];
        const float bhr = b_hh[jr], bhz = b_hh[jz], bhn  = b_hh[jn];
        v8f air = {bir, bir, bir, bir, bir, bir, bir, bir};
        v8f aiz = {biz, biz, biz, biz, biz, biz, biz, biz};
        v8f ain = {bin_, bin_, bin_, bin_, bin_, bin_, bin_, bin_};
        v8f ahr = {bhr, bhr, bhr, bhr, bhr, bhr, bhr, bhr};
        v8f ahz = {bhz, bhz, bhz, bhz, bhz, bhz, bhz, bhz};
        v8f ahn = {bhn, bhn, bhn, bhn, bhn, bhn, bhn, bhn};

        #pragma unroll 2
        for (int k0 = 0; k0 < MEM_DIM; k0 += 4) {      // 172 = 43*4 exactly
            const v2f am = *(const v2f*)(amBase + k0);
            const v2f ah = *(const v2f*)(ahBase + k0);
            air = wmma_f32(am, *(const v2f*)(pir + k0), air);
            aiz = wmma_f32(am, *(const v2f*)(piz + k0), aiz);
            ain = wmma_f32(am, *(const v2f*)(pin + k0), ain);
            ahr = wmma_f32(ah, *(const v2f*)(phr + k0), ahr);
            ahz = wmma_f32(ah, *(const v2f*)(phz + k0), ahz);
            ahn = wmma_f32(ah, *(const v2f*)(phn + k0), ahn);
        }

        // GRU elementwise on the D fragments (identical layouts), then
        // scatter: lane ln -> col j0+ln, vgpr r -> batch row r + 8*hl.
        const int  col    = j0 + ln;
        const bool cvalid = col < MEM_DIM;
        const int  safec  = cvalid ? col : 0;
        #pragma unroll
        for (int r = 0; r < 8; ++r) {
            const int row = r + hl * 8;
            const float hv = hT[row * PADT + safec];
            const float rg = fast_sigmoid(air[r] + ahr[r]);
            const float zg = fast_sigmoid(aiz[r] + ahz[r]);
            const float ng = fast_tanh(ain[r] + rg * ahn[r]);
            const float o  = (1.0f - zg) * ng + zg * hv;
            if (wactive && cvalid && (rb0 + row) < batch) {
                const long long nid = (long long)node_ids[rb0 + row];
                out[nid * (long long)MEM_DIM + col] = o;
            }
        }
        wait_async_all();
        __syncthreads();
    }
}

extern "C" void kernel_launch(void* const* d_in, const int* in_sizes, int n_in,
                              void* d_out, int out_size, void* d_ws, size_t ws_size,
                              hipStream_t stream) {
    const float* node_memories = (const float*)d_in[0];
    const float* X             = (const float*)d_in[1];
    const float* Wr            = (const float*)d_in[2];
    const float* wih           = (const float*)d_in[3];
    const float* whh           = (const float*)d_in[4];
    const float* bih           = (const float*)d_in[5];
    const float* bhh           = (const float*)d_in[6];
    const int*   ids           = (const int*)d_in[7];
    float* out = (float*)d_out;

    const int batch = in_sizes[7];

    // Pass 1: copy-through of the full memory bank (688 MB of the ~1 GB
    // total HBM traffic); D2D memcpy saturates HBM, graph-capture legal.
    hipMemcpyAsync(out, node_memories, (size_t)out_size * sizeof(float),
                   hipMemcpyDeviceToDevice, stream);

    // Pass 2: fused GEMM1 -> GEMM2/3 -> GRU -> scatter.
    const int ntiles = (batch + 15) / 16;                       // 6250
    const int blocks = (ntiles + WAVES_PER_BLK - 1) / WAVES_PER_BLK;
    const size_t shmem = (size_t)SMEM_FLOATS * sizeof(float);   // ~322.5 KB

    hipFuncSetAttribute((const void*)tgn_gru_fused,
                        hipFuncAttributeMaxDynamicSharedMemorySize,
                        (int)shmem);

    tgn_gru_fused<<<blocks, NTHREADS, shmem, stream>>>(
        node_memories, X, Wr, wih, whh, bih, bhh, ids, out, batch);
}